// SpatialAttention_79499844649042
// MI455X (gfx1250) — compile-verified
//
#include <hip/hip_runtime.h>
#include <hip/hip_bf16.h>
#include <math.h>

// ---------------------------------------------------------------------------
// SpatialAttention for MI455X (gfx1250, wave32, WMMA f32 16x16x4).
//
//   t = W @ text + b                      [B, DH, T]
//   s = image^T @ t                       stored transposed: sT [B, T, N]
//   softmax over N (column softmax)       -> colmax/colsum over N per (b,t)
//   out[b,ch,n] = sum_t softmax(s)[n,t] * t[ch,t]
//
// Workspace requirement: ~277 MB (t_mat 4MB + sT 256MB + partials).
//
// CDNA5 paths used: v_wmma_f32_16x16x4_f32 (all three GEMMs),
// global_load_async_to_lds_b128/b32 + s_wait_asynccnt for LDS staging
// (guarded by __has_builtin, with plain-load fallback), global_prefetch.
// ---------------------------------------------------------------------------

#define B_   64
#define D_   256
#define T_   64
#define DH_  256
#define N_   16384
#define NTROWS 128
#define NTILES (N_ / NTROWS)   // 128

typedef __attribute__((ext_vector_type(2))) float v2f;
typedef __attribute__((ext_vector_type(4))) float v4f;
typedef __attribute__((ext_vector_type(8))) float v8f;
typedef __attribute__((ext_vector_type(4))) int   v4i;

#if defined(__has_builtin)
#if __has_builtin(__builtin_amdgcn_global_load_async_to_lds_b128) && \
    __has_builtin(__builtin_amdgcn_global_load_async_to_lds_b32) &&  \
    __has_builtin(__builtin_amdgcn_s_wait_asynccnt)
#define HAVE_ASYNC_LDS 1
#endif
#endif
#ifndef HAVE_ASYNC_LDS
#define HAVE_ASYNC_LDS 0
#endif

#if HAVE_ASYNC_LDS
// Exact param types per hipcc diagnostics:
//   b128: (v4i addrspace(1)*, v4i addrspace(3)*, imm, imm)
//   b32 : (int addrspace(1)*, int addrspace(3)*, imm, imm)
typedef __attribute__((address_space(1))) v4i gv4i_t;
typedef __attribute__((address_space(3))) v4i lv4i_t;
typedef __attribute__((address_space(1))) int gint_t;
typedef __attribute__((address_space(3))) int lint_t;
#endif

// D = A(16x4) * B(4x16) + C, fp32.  8-arg form: (neg_a,A,neg_b,B,c_mod,C,reuse_a,reuse_b)
__device__ __forceinline__ v8f wmma4(v2f a, v2f b, v8f c) {
  return __builtin_amdgcn_wmma_f32_16x16x4_f32(false, a, false, b, (short)0, c,
                                               false, false);
}

// ---------------------------------------------------------------------------
// Kernel 1: t_mat[b][o][t] = sum_d W[o][d] * text[b][d][t] + bias[o]
// One wave per 16x16 output tile; K loop of 64 WMMAs.
// ---------------------------------------------------------------------------
__global__ __launch_bounds__(32) void k_project(const float* __restrict__ W,
                                                const float* __restrict__ text,
                                                const float* __restrict__ bias,
                                                float* __restrict__ t_mat) {
  const int b = blockIdx.z, o0 = blockIdx.y * 16, t0 = blockIdx.x * 16;
  const int lane  = threadIdx.x & 31;
  const int ml    = lane & 15;          // M (A) / N (B,C,D) local index
  const int khalf = (lane >> 4) << 1;   // lanes 16-31 hold K=2,3
  const float* Xb = text + b * D_ * T_;

  v8f acc = {};
  for (int k = 0; k < D_; k += 4) {
    v2f a, bb;
    a.x  = W[(o0 + ml) * D_ + k + khalf];
    a.y  = W[(o0 + ml) * D_ + k + khalf + 1];
    bb.x = Xb[(k + khalf) * T_ + t0 + ml];
    bb.y = Xb[(k + khalf + 1) * T_ + t0 + ml];
    acc  = wmma4(a, bb, acc);
  }
  const int mbase = (lane >> 4) * 8;
#pragma unroll
  for (int r = 0; r < 8; ++r) {
    const int o = o0 + mbase + r;
    t_mat[(b * DH_ + o) * T_ + t0 + ml] = acc[r] + bias[o];
  }
}

// ---------------------------------------------------------------------------
// Kernel 2: sT[b][t][n] = sum_c image[b][c][n] * t_mat[b][c][t]
// Block = 8 waves; each wave owns 16 rows of n x all 64 t (4 WMMA D tiles).
// t_mat staged into LDS in 64-channel chunks via async B128 DMA (ASYNCcnt);
// image A loads are coalesced. Emits per-block column maxima -> pmax.
// ---------------------------------------------------------------------------
#define K2_CH 64
__global__ __launch_bounds__(256) void k_scores(const float* __restrict__ img,
                                                const float* __restrict__ t_mat,
                                                float* __restrict__ sT,
                                                float* __restrict__ pmax) {
  __shared__ __align__(16) float lds_t[K2_CH * T_];  // 16 KB: [chunk-ch][t]
  __shared__ float lds_red[8 * T_];                  // 2 KB

  const int tile = blockIdx.x, b = blockIdx.y;
  const int tid  = threadIdx.x;
  const int wave = tid >> 5, lane = tid & 31;
  const int ml    = lane & 15;
  const int khalf = (lane >> 4) << 1;
  const int nbase = tile * NTROWS + wave * 16;

  const float* tb = t_mat + b * DH_ * T_;
  const float* ib = img + (size_t)b * DH_ * N_;

  v8f acc[4] = {v8f{}, v8f{}, v8f{}, v8f{}};

  for (int c0 = 0; c0 < DH_; c0 += K2_CH) {
    __syncthreads();
#if HAVE_ASYNC_LDS
    {
      // 4096 floats / 256 threads = 4 x B128 per thread; LDS untouched by VALU.
      gv4i_t* gsrc = (gv4i_t*)(tb + c0 * T_);
      lv4i_t* ldst = (lv4i_t*)lds_t;
#pragma unroll
      for (int v = 0; v < 4; ++v) {
        const int e = tid + v * 256;  // v4i element index (16B granules)
        __builtin_amdgcn_global_load_async_to_lds_b128(gsrc + e, ldst + e, 0,
                                                       0);
      }
      __builtin_amdgcn_s_wait_asynccnt(0);
    }
#else
    for (int i = tid; i < K2_CH * T_; i += 256) lds_t[i] = tb[c0 * T_ + i];
#endif
    __syncthreads();

    if (c0 + K2_CH < DH_)  // hint next image chunk into cache (global_prefetch)
      __builtin_prefetch(&ib[(size_t)(c0 + K2_CH) * N_ + nbase + ml], 0, 1);

    for (int kk = 0; kk < K2_CH; kk += 4) {
      const int c = c0 + kk + khalf;
      v2f a;
      a.x = ib[(size_t)c * N_ + nbase + ml];
      a.y = ib[(size_t)(c + 1) * N_ + nbase + ml];
      const int kb = kk + khalf;
#pragma unroll
      for (int tg = 0; tg < 4; ++tg) {
        v2f bb;
        bb.x    = lds_t[kb * T_ + tg * 16 + ml];
        bb.y    = lds_t[(kb + 1) * T_ + tg * 16 + ml];
        acc[tg] = wmma4(a, bb, acc[tg]);
      }
    }
  }

  // Store sT (two B128 per D tile per lane) + per-wave column max.
  const int mbase = (lane >> 4) * 8;
#pragma unroll
  for (int tg = 0; tg < 4; ++tg) {
    const int t = tg * 16 + ml;
    float* dst = sT + ((size_t)(b * T_ + t)) * N_ + nbase + mbase;
    v4f lo = {acc[tg][0], acc[tg][1], acc[tg][2], acc[tg][3]};
    v4f hi = {acc[tg][4], acc[tg][5], acc[tg][6], acc[tg][7]};
    *(v4f*)dst       = lo;
    *(v4f*)(dst + 4) = hi;

    float m = acc[tg][0];
#pragma unroll
    for (int r = 1; r < 8; ++r) m = fmaxf(m, acc[tg][r]);
    m = fmaxf(m, __shfl_xor(m, 16, 32));  // merge the two M-halves (same t)
    if (lane < 16) lds_red[wave * T_ + t] = m;
  }
  __syncthreads();
  if (tid < T_) {
    float m = lds_red[tid];
#pragma unroll
    for (int w = 1; w < 8; ++w) m = fmaxf(m, lds_red[w * T_ + tid]);
    pmax[((size_t)b * NTILES + tile) * T_ + tid] = m;
  }
}

// ---------------------------------------------------------------------------
// Kernel 3: colmax[b][t] = max over tiles of pmax
// ---------------------------------------------------------------------------
__global__ __launch_bounds__(64) void k_colmax(const float* __restrict__ pmax,
                                               float* __restrict__ cmax) {
  const int b = blockIdx.x, t = threadIdx.x;
  float m = -3.402823466e38f;
  for (int i = 0; i < NTILES; ++i)
    m = fmaxf(m, pmax[((size_t)b * NTILES + i) * T_ + t]);
  cmax[b * T_ + t] = m;
}

// ---------------------------------------------------------------------------
// Kernel 4: psum[b][tile][t] = sum_{n in tile} exp(sT[b][t][n] - cmax[b][t])
// Deterministic LDS tree reduction (no float atomics).
// ---------------------------------------------------------------------------
__global__ __launch_bounds__(256) void k_expsum(const float* __restrict__ sT,
                                                const float* __restrict__ cmax,
                                                float* __restrict__ psum) {
  __shared__ float red[256];
  const int tile = blockIdx.x, b = blockIdx.y, tid = threadIdx.x;
  const int t = tid & 63, c = tid >> 6;  // 4 n-segments per t
  const float m = cmax[b * T_ + t];
  const float* src =
      sT + ((size_t)(b * T_ + t)) * N_ + tile * NTROWS + c * 32;
  float s = 0.f;
#pragma unroll 8
  for (int i = 0; i < 32; ++i) s += __expf(src[i] - m);
  red[tid] = s;
  __syncthreads();
  if (tid < 64) {
    float tot = red[tid] + red[tid + 64] + red[tid + 128] + red[tid + 192];
    psum[((size_t)b * NTILES + tile) * T_ + tid] = tot;
  }
}

// ---------------------------------------------------------------------------
// Kernel 5: crcp[b][t] = 1 / sum over tiles of psum
// ---------------------------------------------------------------------------
__global__ __launch_bounds__(64) void k_colrcp(const float* __restrict__ psum,
                                               float* __restrict__ crcp) {
  const int b = blockIdx.x, t = threadIdx.x;
  float s = 0.f;
  for (int i = 0; i < NTILES; ++i)
    s += psum[((size_t)b * NTILES + i) * T_ + t];
  crcp[b * T_ + t] = 1.0f / s;
}

// ---------------------------------------------------------------------------
// Kernel 6: out[b][ch][n] = sum_t P(n,t) * t_mat[b][ch][t],
//           P(n,t) = exp(sT[b][t][n] - cmax) * crcp   (built in the A frag)
// Block = 8 waves: one 16-ch tile, 128 n rows; t^T tile staged in LDS via
// per-element async B32 DMA (transpose gather on the LDS side).
// ---------------------------------------------------------------------------
__global__ __launch_bounds__(256) void k_attend(const float* __restrict__ sT,
                                                const float* __restrict__ t_mat,
                                                const float* __restrict__ cmax,
                                                const float* __restrict__ crcp,
                                                float* __restrict__ out) {
  __shared__ __align__(16) float lds_bt[T_ * 16];  // 4 KB: [t][ch_local]
  const int ch0 = blockIdx.x * 16, ntile = blockIdx.y, b = blockIdx.z;
  const int tid = threadIdx.x, wave = tid >> 5, lane = tid & 31;
  const int ml = lane & 15, khalf = (lane >> 4) << 1, mbase = (lane >> 4) * 8;

#if HAVE_ASYNC_LDS
  {
    // Global side contiguous (i = chl*T + t); LDS side transposed [t][chl].
    gint_t* gsrc = (gint_t*)(t_mat + (size_t)b * DH_ * T_ + ch0 * T_);
    lint_t* ldst = (lint_t*)lds_bt;
#pragma unroll
    for (int v = 0; v < 4; ++v) {
      const int i   = tid + v * 256;        // 0 .. 1023
      const int chl = i >> 6, t = i & 63;   // i = chl*T_ + t
      __builtin_amdgcn_global_load_async_to_lds_b32(gsrc + i,
                                                    ldst + (t * 16 + chl), 0,
                                                    0);
    }
    __builtin_amdgcn_s_wait_asynccnt(0);
  }
#else
  for (int i = tid; i < T_ * 16; i += 256) {
    const int chl = i >> 6, t = i & 63;
    lds_bt[t * 16 + chl] = t_mat[((size_t)b * DH_ + ch0 + chl) * T_ + t];
  }
#endif
  __syncthreads();

  const int n0 = ntile * NTROWS + wave * 16;
  const float* sb = sT + (size_t)b * T_ * N_;
  const float* cm = cmax + b * T_;
  const float* cr = crcp + b * T_;

  v8f acc = {};
  for (int kk = 0; kk < T_; kk += 4) {
    const int klo = kk + khalf;
    v2f a, bb;
    a.x = __expf(sb[(size_t)klo * N_ + n0 + ml] - cm[klo]) * cr[klo];
    a.y = __expf(sb[(size_t)(klo + 1) * N_ + n0 + ml] - cm[klo + 1]) * cr[klo + 1];
    bb.x = lds_bt[klo * 16 + ml];
    bb.y = lds_bt[(klo + 1) * 16 + ml];
    acc = wmma4(a, bb, acc);
  }

  // D[m][chl]: lane holds chl = ml, rows mbase+r -> contiguous in n.
  float* dst = out + ((size_t)b * DH_ + ch0 + ml) * N_ + n0 + mbase;
  v4f lo = {acc[0], acc[1], acc[2], acc[3]};
  v4f hi = {acc[4], acc[5], acc[6], acc[7]};
  *(v4f*)dst       = lo;
  *(v4f*)(dst + 4) = hi;
}

// ---------------------------------------------------------------------------
extern "C" void kernel_launch(void* const* d_in, const int* in_sizes, int n_in,
                              void* d_out, int out_size, void* d_ws,
                              size_t ws_size, hipStream_t stream) {
  const float* text  = (const float*)d_in[0];  // [B, D, T]
  const float* image = (const float*)d_in[1];  // [B, DH, N]
  const float* W     = (const float*)d_in[2];  // [DH, D]
  const float* bias  = (const float*)d_in[3];  // [DH]
  float* out = (float*)d_out;                  // [B, DH, N]

  float* ws    = (float*)d_ws;
  float* t_mat = ws;                                   // B*DH*T   (4 MB)
  float* sT    = t_mat + (size_t)B_ * DH_ * T_;        // B*T*N    (256 MB)
  float* pmax  = sT + (size_t)B_ * T_ * N_;            // B*NTILES*T (2 MB)
  float* psum  = pmax + (size_t)B_ * NTILES * T_;      // B*NTILES*T (2 MB)
  float* cmax  = psum + (size_t)B_ * NTILES * T_;      // B*T
  float* crcp  = cmax + B_ * T_;                       // B*T

  k_project<<<dim3(T_ / 16, DH_ / 16, B_), 32, 0, stream>>>(W, text, bias,
                                                            t_mat);
  k_scores<<<dim3(NTILES, B_), 256, 0, stream>>>(image, t_mat, sT, pmax);
  k_colmax<<<B_, T_, 0, stream>>>(pmax, cmax);
  k_expsum<<<dim3(NTILES, B_), 256, 0, stream>>>(sT, cmax, psum);
  k_colrcp<<<B_, T_, 0, stream>>>(psum, crcp);
  k_attend<<<dim3(DH_ / 16, N_ / NTROWS, B_), 256, 0, stream>>>(
      sT, t_mat, cmax, crcp, out);
}